// GAT_7215545057458
// MI455X (gfx1250) — compile-verified
//
#include <hip/hip_runtime.h>
#include <climits>

#define IN_C  128
#define OUT_C 128   // HEADS * HID
#define NEG_SLOPE 0.2f

typedef __attribute__((ext_vector_type(2))) float v2f;
typedef __attribute__((ext_vector_type(8))) float v8f;

// ---- float <-> order-preserving int (for atomicMax on signed ints) ----
__device__ __forceinline__ int f2oi(float f) {
  int i = __float_as_int(f);
  return i >= 0 ? i : (i ^ 0x7FFFFFFF);
}
__device__ __forceinline__ float oi2f(int i) {
  return __int_as_float(i >= 0 ? i : (i ^ 0x7FFFFFFF));
}

// ---- init: zero output accumulator + denom, set segment-max identity ----
__global__ void gat_init(float* __restrict__ out, float* __restrict__ denom,
                         int* __restrict__ amax, int n128, int n2) {
  int i = blockIdx.x * blockDim.x + threadIdx.x;
  if (i < n128) out[i] = 0.0f;
  if (i < n2) { denom[i] = 0.0f; amax[i] = INT_MIN; }
}

// ---- fused dual GEMM: xl = x@Wl + bl, xr = x@Wr + br, via f32 WMMA ----
// One 256-thread block = 8 waves; wave w owns column tile w (8 * 16 = 128 cols).
// Each wave computes a 16x16 tile for BOTH weight matrices, reusing A regs.
__global__ __launch_bounds__(256) void gat_gemm_wmma(
    const float* __restrict__ x,
    const float* __restrict__ Wl, const float* __restrict__ bl,
    const float* __restrict__ Wr, const float* __restrict__ br,
    float* __restrict__ xl, float* __restrict__ xr, int N) {
  const int lane    = threadIdx.x & 31;
  const int wave    = threadIdx.x >> 5;      // column tile 0..7
  const int tileM   = blockIdx.x;
  const int halfsel = lane >> 4;             // 0 or 1 (K half for A/B)
  const int kb      = halfsel * 2;

  int row = tileM * 16 + (lane & 15);
  if (row >= N) row = N - 1;                 // clamp address (full-tile common case)
  const int col = wave * 16 + (lane & 15);
  const float* xrow = x + (size_t)row * IN_C;

  // C/D layout: vgpr v holds (M = v + 8*halfsel, N = lane&15); bias is per-col.
  v8f accL, accR;
  const float cl = bl[col], cr = br[col];
#pragma unroll
  for (int i = 0; i < 8; ++i) { accL[i] = cl; accR[i] = cr; }

#pragma unroll 4
  for (int k0 = 0; k0 < IN_C; k0 += 4) {
    v2f a, bL, bR;
    // A 16x4: lane = row, VGPR0/1 = K = kb, kb+1
    a[0] = xrow[k0 + kb];
    a[1] = xrow[k0 + kb + 1];
    // B 4x16: lane = col, VGPR0/1 = K = kb, kb+1  (W is [K=128, N=128] row-major)
    bL[0] = Wl[(size_t)(k0 + kb)     * OUT_C + col];
    bL[1] = Wl[(size_t)(k0 + kb + 1) * OUT_C + col];
    bR[0] = Wr[(size_t)(k0 + kb)     * OUT_C + col];
    bR[1] = Wr[(size_t)(k0 + kb + 1) * OUT_C + col];
    accL = __builtin_amdgcn_wmma_f32_16x16x4_f32(false, a, false, bL,
                                                 (short)0, accL, false, false);
    accR = __builtin_amdgcn_wmma_f32_16x16x4_f32(false, a, false, bR,
                                                 (short)0, accR, false, false);
  }

  // Epilogue: tile-uniform guard -> straight-line stores in the common case.
  const int mbase = tileM * 16 + 8 * halfsel;
  float* pl = xl + (size_t)mbase * OUT_C + col;
  float* pr = xr + (size_t)mbase * OUT_C + col;
  if (tileM * 16 + 16 <= N) {
#pragma unroll
    for (int v = 0; v < 8; ++v) {
      pl[(size_t)v * OUT_C] = accL[v];
      pr[(size_t)v * OUT_C] = accR[v];
    }
  } else {
#pragma unroll
    for (int v = 0; v < 8; ++v) {
      if (mbase + v < N) {
        pl[(size_t)v * OUT_C] = accL[v];
        pr[(size_t)v * OUT_C] = accR[v];
      }
    }
  }
}

// ---- per-edge GATv2 score: alpha[e][h] = att[h] . LeakyReLU(xl[src]+xr[dst])
// One wave per edge; lane handles 4 channels (float4). Lanes 0-15 = head 0
// (ch 0-63), lanes 16-31 = head 1 (ch 64-127). Half-wave xor reduction.
__global__ __launch_bounds__(256) void gat_edge_score(
    const float* __restrict__ xl, const float* __restrict__ xr,
    const float* __restrict__ att, const int* __restrict__ src,
    const int* __restrict__ dst, float* __restrict__ alpha,
    int* __restrict__ amax, int E) {
  int e = blockIdx.x * 8 + (threadIdx.x >> 5);
  if (e >= E) return;
  const int lane = threadIdx.x & 31;
  const int s = src[e], d = dst[e];

  float4 a = ((const float4*)(xl + (size_t)s * OUT_C))[lane];
  float4 b = ((const float4*)(xr + (size_t)d * OUT_C))[lane];
  float4 w = ((const float4*)att)[lane];   // att flat [2*64] == channel index

  float v, p = 0.0f;
  v = a.x + b.x; v = v >= 0.0f ? v : NEG_SLOPE * v; p += w.x * v;
  v = a.y + b.y; v = v >= 0.0f ? v : NEG_SLOPE * v; p += w.y * v;
  v = a.z + b.z; v = v >= 0.0f ? v : NEG_SLOPE * v; p += w.z * v;
  v = a.w + b.w; v = v >= 0.0f ? v : NEG_SLOPE * v; p += w.w * v;

#pragma unroll
  for (int off = 8; off; off >>= 1) p += __shfl_xor(p, off, 32);

  if ((lane & 15) == 0) {
    const int h = lane >> 4;
    alpha[(size_t)e * 2 + h] = p;
    atomicMax(&amax[(size_t)d * 2 + h], f2oi(p));
  }
}

// ---- exp(alpha - max) and denom accumulation; one thread per (edge, head) --
__global__ void gat_edge_exp(float* __restrict__ alpha,
                             const int* __restrict__ amax,
                             float* __restrict__ denom,
                             const int* __restrict__ dst, int E) {
  int idx = blockIdx.x * blockDim.x + threadIdx.x;
  if (idx >= 2 * E) return;
  const int e = idx >> 1, h = idx & 1;
  const int d = dst[e];
  const float m = oi2f(amax[(size_t)d * 2 + h]);
  const float v = __expf(alpha[idx] - m);
  alpha[idx] = v;                               // overwrite raw score with exp
  atomicAdd(&denom[(size_t)d * 2 + h], v);
}

// ---- weighted scatter-add: out[dst] += xl[src] * (exp / denom) ----
__global__ __launch_bounds__(256) void gat_aggregate(
    const float* __restrict__ xl, const float* __restrict__ alpha,
    const float* __restrict__ denom, const int* __restrict__ src,
    const int* __restrict__ dst, float* __restrict__ out, int E) {
  int e = blockIdx.x * 8 + (threadIdx.x >> 5);
  if (e >= E) return;
  const int lane = threadIdx.x & 31;
  const int s = src[e], d = dst[e];
  const int h = lane >> 4;
  const float w = alpha[(size_t)e * 2 + h] /
                  (denom[(size_t)d * 2 + h] + 1e-16f);
  float4 a = ((const float4*)(xl + (size_t)s * OUT_C))[lane];
  float* o = out + (size_t)d * OUT_C + lane * 4;
  atomicAdd(o + 0, a.x * w);
  atomicAdd(o + 1, a.y * w);
  atomicAdd(o + 2, a.z * w);
  atomicAdd(o + 3, a.w * w);
}

// ---- finalize: tanh(acc + bias) in place ----
__global__ void gat_finalize(float* __restrict__ out,
                             const float* __restrict__ bias, int n128) {
  int i = blockIdx.x * blockDim.x + threadIdx.x;
  if (i >= n128) return;
  out[i] = tanhf(out[i] + bias[i & (OUT_C - 1)]);
}

extern "C" void kernel_launch(void* const* d_in, const int* in_sizes, int n_in,
                              void* d_out, int out_size, void* d_ws, size_t ws_size,
                              hipStream_t stream) {
  const float* x    = (const float*)d_in[0];
  const int*   ei   = (const int*)d_in[1];
  // d_in[2] = edge_weight (unused by the reference)
  const float* Wl   = (const float*)d_in[3];
  const float* bl   = (const float*)d_in[4];
  const float* Wr   = (const float*)d_in[5];
  const float* br   = (const float*)d_in[6];
  const float* att  = (const float*)d_in[7];
  const float* bias = (const float*)d_in[8];

  const int N = in_sizes[0] / IN_C;
  const int E = in_sizes[1] / 2;
  const int* src = ei;
  const int* dst = ei + E;

  // workspace layout (floats): xl | xr | alpha(2E) | amax(2N ints) | denom(2N)
  float* xl    = (float*)d_ws;
  float* xr    = xl + (size_t)N * OUT_C;
  float* alpha = xr + (size_t)N * OUT_C;
  int*   amax  = (int*)(alpha + (size_t)E * 2);
  float* denom = (float*)(amax + (size_t)N * 2);
  float* out   = (float*)d_out;

  const int n128 = N * OUT_C;
  const int n2   = N * 2;

  gat_init<<<(n128 + 255) / 256, 256, 0, stream>>>(out, denom, amax, n128, n2);
  gat_gemm_wmma<<<(N + 15) / 16, 256, 0, stream>>>(x, Wl, bl, Wr, br, xl, xr, N);
  gat_edge_score<<<(E + 7) / 8, 256, 0, stream>>>(xl, xr, att, src, dst,
                                                  alpha, amax, E);
  gat_edge_exp<<<(2 * E + 255) / 256, 256, 0, stream>>>(alpha, amax, denom,
                                                        dst, E);
  gat_aggregate<<<(E + 7) / 8, 256, 0, stream>>>(xl, alpha, denom, src, dst,
                                                 out, E);
  gat_finalize<<<(n128 + 255) / 256, 256, 0, stream>>>(out, bias, n128);
}